// Sparsify1D_7627861918121
// MI455X (gfx1250) — compile-verified
//
#include <hip/hip_runtime.h>
#include <stdint.h>

// Sparsify1D: per-row top-K (K=64) threshold, then mask-multiply.
// Memory-bound: 1 GB total traffic -> ~44us floor at 23.3 TB/s.
// One 256-thread block per row; async-DMA the 64KB row into LDS
// (GLOBAL_LOAD_ASYNC_TO_LDS_B128 / ASYNCcnt), radix-select the 64th largest
// entirely in LDS (4 x 8-bit MSB-first passes, bank-interleaved histograms),
// then one non-temporal masked float4 write-back. HBM touched once each way.

#define COLS   16384
#define KSEL   64
#define TPB    256
#define NHIST  8                   // histogram copies
#define HPAD   257                 // 256 bins + 1 pad word -> bank (c+b)%64
#define F4PT   (COLS / 4 / TPB)    // 16 float4 loads per thread
#define EPT    (COLS / TPB)        // 64 elements per thread

// Exact parameter types for __builtin_amdgcn_global_load_async_to_lds_b128:
// (int4 AS1*, int4 AS3*, imm offset, imm cpol)
typedef int v4i __attribute__((vector_size(16)));
typedef v4i __attribute__((address_space(1)))* v4i_gp;  // global (__device__)
typedef v4i __attribute__((address_space(3)))* v4i_lp;  // LDS (__shared__)
typedef float v4f __attribute__((ext_vector_type(4)));

__global__ __launch_bounds__(TPB)
void sparsify_topk_rowsel(const float* __restrict__ x, float* __restrict__ out) {
  __shared__ float    s_row[COLS];            // 64 KB row tile
  __shared__ unsigned s_hist[NHIST * HPAD];   // ~8 KB interleaved histograms
  __shared__ unsigned s_scan[256];            // suffix sums
  __shared__ unsigned s_bcast[2];             // {prefix, remaining}

  const unsigned tid     = threadIdx.x;
  const size_t   rowBase = (size_t)blockIdx.x * (size_t)COLS;

  // ---------------- stage row into LDS via CDNA5 async DMA ----------------
  {
    const uint64_t gaddr = (uint64_t)(uintptr_t)(x + rowBase) + (uint64_t)tid * 16u;
    const uint32_t laddr = (uint32_t)(uintptr_t)&s_row[0] + tid * 16u;
#pragma unroll
    for (int j = 0; j < F4PT; ++j) {
      const uint64_t ga = gaddr + (uint64_t)j * (TPB * 16);
      const uint32_t la = laddr + (uint32_t)j * (TPB * 16);
#if __has_builtin(__builtin_amdgcn_global_load_async_to_lds_b128)
      __builtin_amdgcn_global_load_async_to_lds_b128(
          (v4i_gp)(uintptr_t)ga, (v4i_lp)(uintptr_t)la, 0, 0);
#else
      asm volatile("global_load_async_to_lds_b128 %0, %1, off"
                   :: "v"(la), "v"(ga) : "memory");
#endif
    }
  }
#if __has_builtin(__builtin_amdgcn_s_wait_asynccnt)
  __builtin_amdgcn_s_wait_asynccnt(0);
#else
  asm volatile("s_wait_asynccnt 0x0" ::: "memory");
#endif
  __syncthreads();   // all waves' DMA complete -> full row visible in LDS

  // ---------------- 4-pass MSB-first radix select for rank K --------------
  // key(f) is a strictly monotone uint32 mapping of fp32.
  unsigned prefix    = 0u;
  unsigned remaining = KSEL;

  // Lane-interleaved, padded histogram copy: 32 lanes hitting the same hot
  // bin spread across 8 copies in 8 distinct LDS banks (~4-way instead of
  // 32-way serialization on clustered-exponent data).
  unsigned* myHist = &s_hist[(tid & (NHIST - 1)) * HPAD];

#pragma unroll
  for (int pass = 0; pass < 4; ++pass) {
    const int      shift    = 24 - pass * 8;
    const unsigned maskHigh = (pass == 0) ? 0u : (0xFFFFFFFFu << ((24 - pass * 8) + 8));

    for (int i = (int)tid; i < NHIST * HPAD; i += TPB) s_hist[i] = 0u;
    __syncthreads();

#pragma unroll
    for (int j = 0; j < EPT; ++j) {
      const float    f   = s_row[tid + TPB * j];   // bank-conflict free
      const unsigned u   = __float_as_uint(f);
      const unsigned key = u ^ (((unsigned)((int)u >> 31)) | 0x80000000u);
      if (((key ^ prefix) & maskHigh) == 0u)
        atomicAdd(&myHist[(key >> shift) & 0xFFu], 1u);
    }
    __syncthreads();

    // reduce 8 histogram copies -> s_scan (one bin per thread)
    unsigned total = 0u;
#pragma unroll
    for (int c = 0; c < NHIST; ++c) total += s_hist[c * HPAD + tid];
    s_scan[tid] = total;
    __syncthreads();

    // inclusive suffix-sum: s_scan[b] = #keys with digit >= b (in prefix set)
    for (int off = 1; off < 256; off <<= 1) {
      const unsigned v = s_scan[tid];
      const unsigned a = (tid + off < 256u) ? s_scan[tid + off] : 0u;
      __syncthreads();
      s_scan[tid] = v + a;
      __syncthreads();
    }

    // unique bin b with suffix[b] >= remaining > suffix[b+1]
    const unsigned geq = s_scan[tid];
    const unsigned gt  = (tid == 255u) ? 0u : s_scan[tid + 1];
    if (geq >= remaining && gt < remaining) {
      s_bcast[0] = prefix | (tid << shift);
      s_bcast[1] = remaining - gt;
    }
    __syncthreads();
    prefix    = s_bcast[0];
    remaining = s_bcast[1];
  }

  // prefix == key of the K-th largest; invert the monotone map and compare
  // as floats so tie semantics match the reference exactly.
  const unsigned tu = (prefix & 0x80000000u) ? (prefix ^ 0x80000000u) : ~prefix;
  const float thresh = __uint_as_float(tu);

  // -------- masked write-back: single streaming (non-temporal) pass -------
  const v4f* s4 = (const v4f*)s_row;
  v4f*       o4 = (v4f*)(out + rowBase);
#pragma unroll
  for (int j = 0; j < F4PT; ++j) {
    v4f f = s4[tid + TPB * j];
    f.x = (f.x >= thresh) ? f.x : 0.0f;
    f.y = (f.y >= thresh) ? f.y : 0.0f;
    f.z = (f.z >= thresh) ? f.z : 0.0f;
    f.w = (f.w >= thresh) ? f.w : 0.0f;
    __builtin_nontemporal_store(f, &o4[tid + TPB * j]);
  }
}

extern "C" void kernel_launch(void* const* d_in, const int* in_sizes, int n_in,
                              void* d_out, int out_size, void* d_ws, size_t ws_size,
                              hipStream_t stream) {
  (void)n_in; (void)out_size; (void)d_ws; (void)ws_size;
  const float* x   = (const float*)d_in[0];
  float*       out = (float*)d_out;
  const int rows = in_sizes[0] / COLS;   // 8192 for the reference shapes
  sparsify_topk_rowsel<<<dim3(rows), dim3(TPB), 0, stream>>>(x, out);
}